// GatedGIN_73658689126816
// MI455X (gfx1250) — compile-verified
//
#include <hip/hip_runtime.h>
#include <hip/hip_bf16.h>

#define GN 100000
#define GE 1600000
#define GH 128
#define NH (GN * GH)          // 12,800,000 floats per activation buffer
#define LDS_STRIDE 132        // 128 + 4 pad -> conflict-free WMMA A reads

typedef __attribute__((ext_vector_type(2))) float v2f;
typedef __attribute__((ext_vector_type(8))) float v8f;
typedef __attribute__((ext_vector_type(4))) int   v4i;

#define AS1 __attribute__((address_space(1)))
#define AS3 __attribute__((address_space(3)))

#if __has_builtin(__builtin_amdgcn_global_load_async_to_lds_b128) && \
    __has_builtin(__builtin_amdgcn_s_wait_asynccnt)
#define HAVE_ASYNC_LDS 1
#endif

// Copy 16 bytes global -> LDS. CDNA5 path: ASYNCcnt-tracked direct-to-LDS DMA.
__device__ __forceinline__ void async_copy_f4(const float* __restrict__ gsrc,
                                              float* __restrict__ ldst) {
#ifdef HAVE_ASYNC_LDS
    float* g = const_cast<float*>(gsrc);
    __builtin_amdgcn_global_load_async_to_lds_b128(
        (AS1 v4i*)g, (AS3 v4i*)ldst, 0, 0);
#else
    float4 v = *(const float4*)gsrc;
    ldst[0] = v.x; ldst[1] = v.y; ldst[2] = v.z; ldst[3] = v.w;
#endif
}

__device__ __forceinline__ void async_wait_all() {
#ifdef HAVE_ASYNC_LDS
    __builtin_amdgcn_s_wait_asynccnt(0);
#endif
}

__device__ __forceinline__ float sigmoidf_(float x) {
    return 1.0f / (1.0f + __expf(-x));
}

// ---------------------------------------------------------------------------
// Weight packing: emit k-pair interleaved layout Wp[k/2][col][2] so each WMMA
// B-fragment (B[k0][col], B[k0+1][col]) is ONE coalesced global_load_b64.
// ---------------------------------------------------------------------------

// src: [128,128] row-major, B[k][j] = src[k][j]   (W_first / W_lin1)
__global__ void pack_direct_128(const float* __restrict__ src, float* __restrict__ dst) {
    int idx = blockIdx.x * blockDim.x + threadIdx.x;   // < 16384
    int t = idx >> 8;            // k-pair 0..63
    int j = (idx >> 1) & 127;    // col
    int u = idx & 1;
    dst[idx] = src[(2 * t + u) * GH + j];
}

// src: [384,128] torch layout, B[k][jg] = src[jg][k]  (W_ih / W_hh)
__global__ void pack_tr_384(const float* __restrict__ src, float* __restrict__ dst) {
    int idx = blockIdx.x * blockDim.x + threadIdx.x;   // < 49152
    int t   = idx / 768;
    int rem = idx - t * 768;
    int j   = rem >> 1;          // gate-col 0..383
    int u   = rem & 1;
    dst[idx] = src[j * GH + 2 * t + u];
}

// ---------------------------------------------------------------------------
// Zero-fill (scatter target must start at 0; harness poisons ws with 0xAA).
// ---------------------------------------------------------------------------
__global__ void zero_f4(float4* __restrict__ p, int n4) {
    int stride = gridDim.x * blockDim.x;
    for (int i = blockIdx.x * blockDim.x + threadIdx.x; i < n4; i += stride)
        p[i] = make_float4(0.f, 0.f, 0.f, 0.f);
}

// ---------------------------------------------------------------------------
// out[N,128] = X[N,128] @ W[128,128] + b   — f32 WMMA 16x16x4
// grid.x = N/32, block = 512 (16 waves = 2 row tiles x 8 col tiles).
// ---------------------------------------------------------------------------
__global__ __launch_bounds__(512) void gemm_bias_128(
    const float* __restrict__ X, const float2* __restrict__ Wp,
    const float* __restrict__ bias, float* __restrict__ out) {
    __shared__ float ldsA[32 * LDS_STRIDE];

    const int tid = threadIdx.x;
    const int block_row = blockIdx.x * 32;

    // Stage A tile: 32 rows x 128 f32 (1024 float4, 512 threads x 2)
#pragma unroll
    for (int i = 0; i < 2; ++i) {
        int idx = tid + i * 512;
        int r   = idx >> 5;               // 32 float4 per row
        int c4  = idx & 31;
        async_copy_f4(X + (size_t)(block_row + r) * GH + c4 * 4,
                      &ldsA[r * LDS_STRIDE + c4 * 4]);
    }
    async_wait_all();
    __syncthreads();

    const int wave = tid >> 5;
    const int lane = tid & 31;
    const int m    = lane & 15;
    const int half = lane >> 4;
    const int rloc = (wave >> 3) * 16;            // this wave's row tile in LDS
    const int row_base = block_row + rloc;
    const int col  = (wave & 7) * 16 + m;

    v8f acc;
    const float bv = bias[col];
#pragma unroll
    for (int r = 0; r < 8; ++r) acc[r] = bv;

#pragma unroll
    for (int k = 0; k < GH; k += 4) {
        const int k0 = k + 2 * half;
        v2f a, b;
        a[0] = ldsA[(rloc + m) * LDS_STRIDE + k0];
        a[1] = ldsA[(rloc + m) * LDS_STRIDE + k0 + 1];
        const float2 b2 = Wp[(size_t)(k0 >> 1) * GH + col];   // b64, coalesced
        b[0] = b2.x; b[1] = b2.y;
        acc = __builtin_amdgcn_wmma_f32_16x16x4_f32(false, a, false, b,
                                                    (short)0, acc, false, false);
    }

#pragma unroll
    for (int r = 0; r < 8; ++r)
        out[(size_t)(row_base + half * 8 + r) * GH + col] = acc[r];
}

// ---------------------------------------------------------------------------
// Gather-scale-scatter conv: out[dst] += x[src] * w. One wave per edge.
// x and out both fit in the 192MB L2, so this is L2-atomic bound.
// ---------------------------------------------------------------------------
__global__ __launch_bounds__(256) void gated_conv(
    const float* __restrict__ x, const int* __restrict__ src,
    const int* __restrict__ dst, const float* __restrict__ ew,
    float* __restrict__ out, int nedges) {
    const int gw     = (blockIdx.x * blockDim.x + threadIdx.x) >> 5;
    const int lane   = threadIdx.x & 31;
    const int nwaves = (gridDim.x * blockDim.x) >> 5;
    for (int e = gw; e < nedges; e += nwaves) {
        const int   s = src[e];
        const int   d = dst[e];
        const float w = ew[e];
        float4 v = ((const float4*)(x + (size_t)s * GH))[lane];
        float* o = out + (size_t)d * GH + lane * 4;
        unsafeAtomicAdd(o + 0, v.x * w);
        unsafeAtomicAdd(o + 1, v.y * w);
        unsafeAtomicAdd(o + 2, v.z * w);
        unsafeAtomicAdd(o + 3, v.w * w);
    }
}

// ---------------------------------------------------------------------------
// Fused GRU cell: out = GRU(inp, h). Wp_* are packed [64][384][2] float2.
// Each wave accumulates 6 gate tiles (i_r,i_z,i_n,h_r,h_z,h_n) with WMMA,
// then applies gating; [N,384] intermediates never hit memory.
// grid.x = N/32, block = 512 (16 waves = 2 row tiles x 8 col tiles).
// ---------------------------------------------------------------------------
__global__ __launch_bounds__(512) void gru_cell_128(
    const float* __restrict__ inp, const float* __restrict__ hid,
    const float2* __restrict__ Wp_ih, const float* __restrict__ b_ih,
    const float2* __restrict__ Wp_hh, const float* __restrict__ b_hh,
    float* __restrict__ out) {
    __shared__ float ldsI[32 * LDS_STRIDE];
    __shared__ float ldsH[32 * LDS_STRIDE];

    const int tid = threadIdx.x;
    const int block_row = blockIdx.x * 32;

#pragma unroll
    for (int i = 0; i < 2; ++i) {
        int idx = tid + i * 512;
        int r   = idx >> 5;
        int c4  = idx & 31;
        const size_t go = (size_t)(block_row + r) * GH + c4 * 4;
        const int    lo = r * LDS_STRIDE + c4 * 4;
        async_copy_f4(inp + go, &ldsI[lo]);
        async_copy_f4(hid + go, &ldsH[lo]);
    }
    async_wait_all();
    __syncthreads();

    const int wave = tid >> 5;
    const int lane = tid & 31;
    const int m    = lane & 15;
    const int half = lane >> 4;
    const int rloc = (wave >> 3) * 16;
    const int row_base = block_row + rloc;
    const int col  = (wave & 7) * 16 + m;   // 0..127

    v8f ir, iz, in_, hr, hz, hn;
    {
        const float bir = b_ih[col], biz = b_ih[GH + col], bin = b_ih[2 * GH + col];
        const float bhr = b_hh[col], bhz = b_hh[GH + col], bhn = b_hh[2 * GH + col];
#pragma unroll
        for (int r = 0; r < 8; ++r) {
            ir[r] = bir; iz[r] = biz; in_[r] = bin;
            hr[r] = bhr; hz[r] = bhz; hn[r]  = bhn;
        }
    }

#pragma unroll
    for (int k = 0; k < GH; k += 4) {
        const int k0 = k + 2 * half;
        v2f ai, ah;
        ai[0] = ldsI[(rloc + m) * LDS_STRIDE + k0];
        ai[1] = ldsI[(rloc + m) * LDS_STRIDE + k0 + 1];
        ah[0] = ldsH[(rloc + m) * LDS_STRIDE + k0];
        ah[1] = ldsH[(rloc + m) * LDS_STRIDE + k0 + 1];

        const size_t tb = (size_t)(k0 >> 1) * 384;
        float2 p;
        v2f b;
        p = Wp_ih[tb + col];            b[0] = p.x; b[1] = p.y;
        ir  = __builtin_amdgcn_wmma_f32_16x16x4_f32(false, ai, false, b, (short)0, ir,  false, false);
        p = Wp_ih[tb + GH + col];       b[0] = p.x; b[1] = p.y;
        iz  = __builtin_amdgcn_wmma_f32_16x16x4_f32(false, ai, false, b, (short)0, iz,  false, false);
        p = Wp_ih[tb + 2 * GH + col];   b[0] = p.x; b[1] = p.y;
        in_ = __builtin_amdgcn_wmma_f32_16x16x4_f32(false, ai, false, b, (short)0, in_, false, false);
        p = Wp_hh[tb + col];            b[0] = p.x; b[1] = p.y;
        hr  = __builtin_amdgcn_wmma_f32_16x16x4_f32(false, ah, false, b, (short)0, hr,  false, false);
        p = Wp_hh[tb + GH + col];       b[0] = p.x; b[1] = p.y;
        hz  = __builtin_amdgcn_wmma_f32_16x16x4_f32(false, ah, false, b, (short)0, hz,  false, false);
        p = Wp_hh[tb + 2 * GH + col];   b[0] = p.x; b[1] = p.y;
        hn  = __builtin_amdgcn_wmma_f32_16x16x4_f32(false, ah, false, b, (short)0, hn,  false, false);
    }

#pragma unroll
    for (int r = 0; r < 8; ++r) {
        const float rg = sigmoidf_(ir[r] + hr[r]);
        const float zg = sigmoidf_(iz[r] + hz[r]);
        const float ng = tanhf(in_[r] + rg * hn[r]);
        const float hv = ldsH[(rloc + half * 8 + r) * LDS_STRIDE + col];
        out[(size_t)(row_base + half * 8 + r) * GH + col] = (1.0f - zg) * ng + zg * hv;
    }
}

// ---------------------------------------------------------------------------
// logits = x @ W_out[128,2] + b_out ; log_softmax. One wave per node.
// ---------------------------------------------------------------------------
__global__ __launch_bounds__(256) void out_logsoftmax(
    const float* __restrict__ x, const float* __restrict__ Wo,
    const float* __restrict__ bo, float* __restrict__ out, int nrows) {
    const int gw     = (blockIdx.x * blockDim.x + threadIdx.x) >> 5;
    const int lane   = threadIdx.x & 31;
    const int nwaves = (gridDim.x * blockDim.x) >> 5;
    for (int n = gw; n < nrows; n += nwaves) {
        float4 v = ((const float4*)(x + (size_t)n * GH))[lane];
        const int kb = lane * 4;
        float p0 = v.x * Wo[(kb + 0) * 2 + 0] + v.y * Wo[(kb + 1) * 2 + 0] +
                   v.z * Wo[(kb + 2) * 2 + 0] + v.w * Wo[(kb + 3) * 2 + 0];
        float p1 = v.x * Wo[(kb + 0) * 2 + 1] + v.y * Wo[(kb + 1) * 2 + 1] +
                   v.z * Wo[(kb + 2) * 2 + 1] + v.w * Wo[(kb + 3) * 2 + 1];
#pragma unroll
        for (int off = 16; off > 0; off >>= 1) {
            p0 += __shfl_down(p0, off, 32);
            p1 += __shfl_down(p1, off, 32);
        }
        if (lane == 0) {
            const float l0 = p0 + bo[0];
            const float l1 = p1 + bo[1];
            const float mx  = fmaxf(l0, l1);
            const float lse = mx + logf(expf(l0 - mx) + expf(l1 - mx));
            out[(size_t)n * 2 + 0] = l0 - lse;
            out[(size_t)n * 2 + 1] = l1 - lse;
        }
    }
}

// ---------------------------------------------------------------------------
extern "C" void kernel_launch(void* const* d_in, const int* in_sizes, int n_in,
                              void* d_out, int out_size, void* d_ws, size_t ws_size,
                              hipStream_t stream) {
    const float* x       = (const float*)d_in[0];
    const int*   ei      = (const int*)d_in[1];
    const float* ew      = (const float*)d_in[2];
    const float* W_first = (const float*)d_in[3];
    const float* b_first = (const float*)d_in[4];
    const float* W_ih    = (const float*)d_in[5];
    const float* b_ih    = (const float*)d_in[6];
    const float* W_hh    = (const float*)d_in[7];
    const float* b_hh    = (const float*)d_in[8];
    const float* W_lin1  = (const float*)d_in[9];
    const float* b_lin1  = (const float*)d_in[10];
    const float* W_out   = (const float*)d_in[11];
    const float* b_out   = (const float*)d_in[12];

    const int* src = ei;
    const int* dst = ei + GE;

    float* bufH = (float*)d_ws;              // first_lin output == GRU hidden h
    float* bufB = bufH + (size_t)NH;
    float* bufC = bufB + (size_t)NH;
    float* WpF  = bufC + (size_t)NH;         // packed W_first   [64][128][2]
    float* WpL  = WpF + 16384;               // packed W_lin1    [64][128][2]
    float* WpIH = WpL + 16384;               // packed W_ih^T    [64][384][2]
    float* WpHH = WpIH + 49152;              // packed W_hh^T    [64][384][2]

    const int rowBlocks  = GN / 32;          // 3125
    const int convBlocks = 2048;

    // 0. pack weights into k-pair interleaved WMMA-B layout
    pack_direct_128<<<64, 256, 0, stream>>>(W_first, WpF);
    pack_direct_128<<<64, 256, 0, stream>>>(W_lin1, WpL);
    pack_tr_384<<<192, 256, 0, stream>>>(W_ih, WpIH);
    pack_tr_384<<<192, 256, 0, stream>>>(W_hh, WpHH);

    // 1. h = x @ W_first + b_first
    gemm_bias_128<<<rowBlocks, 512, 0, stream>>>(x, (const float2*)WpF, b_first, bufH);

    // 2. conv1 = scatter_add(h[src] * w, dst)
    zero_f4<<<convBlocks, 256, 0, stream>>>((float4*)bufB, NH / 4);
    gated_conv<<<convBlocks, 256, 0, stream>>>(bufH, src, dst, ew, bufB, GE);

    // 3. x = GRU(conv1, h)
    gru_cell_128<<<rowBlocks, 512, 0, stream>>>(bufB, bufH, (const float2*)WpIH, b_ih,
                                                (const float2*)WpHH, b_hh, bufC);

    // 4. x = x @ W_lin1 + b_lin1
    gemm_bias_128<<<rowBlocks, 512, 0, stream>>>(bufC, (const float2*)WpL, b_lin1, bufB);

    // 5. conv2
    zero_f4<<<convBlocks, 256, 0, stream>>>((float4*)bufC, NH / 4);
    gated_conv<<<convBlocks, 256, 0, stream>>>(bufB, src, dst, ew, bufC, GE);

    // 6. x = GRU(conv2, h)
    gru_cell_128<<<rowBlocks, 512, 0, stream>>>(bufC, bufH, (const float2*)WpIH, b_ih,
                                                (const float2*)WpHH, b_hh, bufB);

    // 7. logits + log_softmax
    out_logsoftmax<<<convBlocks, 256, 0, stream>>>(bufB, W_out, b_out, (float*)d_out, GN);
}